// Trip_CIA_4329327035057
// MI455X (gfx1250) — compile-verified
//
#include <hip/hip_runtime.h>
#include <math.h>

typedef float v2f __attribute__((ext_vector_type(2)));
typedef float v8f __attribute__((ext_vector_type(8)));

#define BB 16
#define NN 80
#define DD 256
#define NSUP 5
#define NQRY 75
#define KS 13   // K1
#define KQ 2    // K2
#define MARGIN 0.2f

// ---------------------------------------------------------------------------
// Kernel 1: qk = inpt @ Wqk + bqk  (per batch 80x512 = 80x256 * 256x512)
// one wave per 16x16 output tile, f32 WMMA 16x16x4.
// grid = 16 * 5 * 32 blocks of 32 threads.
// ---------------------------------------------------------------------------
__global__ void tcia_qk_wmma(const float* __restrict__ inpt,
                             const float* __restrict__ Wqk,
                             const float* __restrict__ bqk,
                             float* __restrict__ q,
                             float* __restrict__ k) {
  int job = blockIdx.x;
  int b  = job / 160;
  int t  = job % 160;
  int mt = t / 32;          // 0..4   (rows 0..79)
  int nt = t % 32;          // 0..31  (cols 0..511)
  int lane = threadIdx.x;
  int lo = lane & 15;
  int hi = lane >> 4;       // 0 or 1

  const float* A = inpt + ((size_t)b * NN + (mt * 16 + lo)) * DD;
  int n = nt * 16 + lo;

  v8f c = {0.f, 0.f, 0.f, 0.f, 0.f, 0.f, 0.f, 0.f};
  for (int kk = 0; kk < DD; kk += 4) {
    v2f a, w;
    a.x = A[kk + 2 * hi];
    a.y = A[kk + 2 * hi + 1];
    w.x = Wqk[(size_t)(kk + 2 * hi) * 512 + n];
    w.y = Wqk[(size_t)(kk + 2 * hi + 1) * 512 + n];
    c = __builtin_amdgcn_wmma_f32_16x16x4_f32(false, a, false, w,
                                              (short)0, c, false, false);
  }
  float bias = bqk[n];
  for (int v = 0; v < 8; ++v) {
    int m = mt * 16 + v + hi * 8;
    float val = c[v] + bias;
    size_t base = ((size_t)b * NN + m) * DD;
    if (n < DD) q[base + n]        = val;
    else        k[base + (n - DD)] = val;
  }
}

// ---------------------------------------------------------------------------
// Kernel 2: feat[b,n,x] = sum_d inpt[b,n,d] * softmax_n(q[b,n,d]*k[b,n,x])
//                         + inpt[b,n,x]
// block = (b, x): 256 threads, thread = d. Online softmax over n per d,
// then chunked LDS tree-reduction over d.
// ---------------------------------------------------------------------------
__global__ void tcia_attn_feat(const float* __restrict__ inpt,
                               const float* __restrict__ q,
                               const float* __restrict__ k,
                               float* __restrict__ feat) {
  __shared__ float ksh[NN];
  __shared__ float inx[NN];
  __shared__ float acc[NN];
  __shared__ float red[256 * 16];

  int b = blockIdx.x;
  int x = blockIdx.y;
  int d = threadIdx.x;
  const float* qb = q    + (size_t)b * NN * DD;
  const float* kb = k    + (size_t)b * NN * DD;
  const float* ib = inpt + (size_t)b * NN * DD;

  if (d < NN) {
    ksh[d] = kb[(size_t)d * DD + x];
    inx[d] = ib[(size_t)d * DD + x];
  }
  __syncthreads();

  // pass 1: online softmax stats over n for this d
  float mx = -3.402823466e38f, sm = 0.f;
  for (int n = 0; n < NN; ++n) {
    float s = qb[(size_t)n * DD + d] * ksh[n];
    if (s > mx) { sm = sm * __expf(mx - s) + 1.f; mx = s; }
    else        { sm += __expf(s - mx); }
  }
  float inv = 1.f / sm;

  // pass 2: weighted contributions, reduced over d in chunks of 16 n's
  for (int n0 = 0; n0 < NN; n0 += 16) {
    for (int j = 0; j < 16; ++j) {
      int n = n0 + j;
      float w = __expf(qb[(size_t)n * DD + d] * ksh[n] - mx) * inv;
      red[d * 16 + j] = ib[(size_t)n * DD + d] * w;
    }
    __syncthreads();
    for (int s = 128; s >= 1; s >>= 1) {
      if (d < s) {
        for (int j = 0; j < 16; ++j)
          red[d * 16 + j] += red[(d + s) * 16 + j];
      }
      __syncthreads();
    }
    if (d < 16) acc[n0 + d] = red[d];
    __syncthreads();
  }

  if (d < NN)
    feat[((size_t)b * NN + d) * DD + x] = acc[d] + inx[d];
}

// ---------------------------------------------------------------------------
// Kernel 3: row L2 norms (or squared norms). grid = 16*80, block = 256.
// ---------------------------------------------------------------------------
__global__ void tcia_rownorm(const float* __restrict__ X,
                             float* __restrict__ out, int do_sqrt) {
  __shared__ float red[256];
  int row = blockIdx.x;
  int t = threadIdx.x;
  float v = X[(size_t)row * DD + t];
  red[t] = v * v;
  __syncthreads();
  for (int s = 128; s >= 1; s >>= 1) {
    if (t < s) red[t] += red[t + s];
    __syncthreads();
  }
  if (t == 0) out[row] = do_sqrt ? sqrtf(red[0]) : red[0];
}

// ---------------------------------------------------------------------------
// Kernel 4: dist[b,s,qy] = -cos(support_s, query_qy) via f32 WMMA.
// grid = 16*5 blocks of 32 (one wave per 16-wide query tile).
// ---------------------------------------------------------------------------
__global__ void tcia_cos_wmma(const float* __restrict__ feat,
                              const float* __restrict__ norms,
                              float* __restrict__ dist) {
  int b  = blockIdx.x / 5;
  int nt = blockIdx.x % 5;
  int lane = threadIdx.x;
  int lo = lane & 15;
  int hi = lane >> 4;
  const float* F = feat + (size_t)b * NN * DD;

  int qy = nt * 16 + lo;
  int rq = 5 + qy; if (rq > NN - 1) rq = NN - 1;   // clamp padded lanes

  v8f c = {0.f, 0.f, 0.f, 0.f, 0.f, 0.f, 0.f, 0.f};
  for (int kk = 0; kk < DD; kk += 4) {
    v2f a, w;
    a.x = F[(size_t)lo * DD + kk + 2 * hi];
    a.y = F[(size_t)lo * DD + kk + 2 * hi + 1];
    w.x = F[(size_t)rq * DD + kk + 2 * hi];
    w.y = F[(size_t)rq * DD + kk + 2 * hi + 1];
    c = __builtin_amdgcn_wmma_f32_16x16x4_f32(false, a, false, w,
                                              (short)0, c, false, false);
  }
  for (int v = 0; v < 8; ++v) {
    int m = v + hi * 8;
    if (m < NSUP && qy < NQRY) {
      float ns = norms[b * NN + m];
      float nq = norms[b * NN + 5 + qy];
      float pn = fmaxf(ns * nq, 1e-6f);
      dist[((size_t)b * NSUP + m) * NQRY + qy] = -(c[v] / pn);
    }
  }
}

// ---------------------------------------------------------------------------
// Kernel 5: support top-13 + MLP softmax gating -> feature[:, :5, :]
// grid = 16*5, block = 256 (thread = feature dim d).
// ---------------------------------------------------------------------------
__global__ void tcia_support_mlp(const float* __restrict__ feat,
                                 const float* __restrict__ dist,
                                 const float* __restrict__ Ws1,
                                 const float* __restrict__ bs1,
                                 const float* __restrict__ Ws2,
                                 const float* __restrict__ bs2,
                                 float* __restrict__ feature) {
  __shared__ int idx[KS];
  int b = blockIdx.x / NSUP;
  int s = blockIdx.x % NSUP;
  int d = threadIdx.x;

  if (d == 0) {
    float dl[NQRY];
    const float* dr = dist + ((size_t)b * NSUP + s) * NQRY;
    for (int j = 0; j < NQRY; ++j) dl[j] = dr[j];
    for (int r = 0; r < KS; ++r) {
      float best = 3.402823466e38f; int bi = 0;
      for (int j = 0; j < NQRY; ++j)
        if (dl[j] < best) { best = dl[j]; bi = j; }   // stable: first min
      idx[r] = bi;
      dl[bi] = 3.402823466e38f;
    }
  }
  __syncthreads();

  const float* F = feat + (size_t)b * NN * DD;
  float x[KS + 1];
  x[0] = F[(size_t)s * DD + d];
  for (int r = 0; r < KS; ++r)
    x[1 + r] = F[(size_t)(5 + idx[r]) * DD + d];

  float h[16];
  for (int u = 0; u < 16; ++u) {
    float a = bs1[u];
    for (int j = 0; j < KS + 1; ++j) a += x[j] * Ws1[j * 16 + u];
    h[u] = fmaxf(a, 0.f);
  }
  float lg[KS + 1];
  float mx = -3.402823466e38f;
  for (int o = 0; o < KS + 1; ++o) {
    float a = bs2[o];
    for (int u = 0; u < 16; ++u) a += h[u] * Ws2[u * (KS + 1) + o];
    lg[o] = a; mx = fmaxf(mx, a);
  }
  float sum = 0.f;
  for (int o = 0; o < KS + 1; ++o) { lg[o] = __expf(lg[o] - mx); sum += lg[o]; }
  float out = 0.f;
  for (int o = 0; o < KS + 1; ++o) out += x[o] * (lg[o] / sum);

  feature[((size_t)b * NN + s) * DD + d] = out;   // protos (N_SHOT==1)
}

// ---------------------------------------------------------------------------
// Kernel 6: query top-2 + MLP softmax gating -> feature[:, 5:, :]
// grid = 16*75, block = 256.
// ---------------------------------------------------------------------------
__global__ void tcia_query_mlp(const float* __restrict__ feat,
                               const float* __restrict__ dist,
                               const float* __restrict__ Wq1,
                               const float* __restrict__ bq1,
                               const float* __restrict__ Wq2,
                               const float* __restrict__ bq2,
                               float* __restrict__ feature) {
  __shared__ int idx[KQ];
  int b  = blockIdx.x / NQRY;
  int qy = blockIdx.x % NQRY;
  int d  = threadIdx.x;

  if (d == 0) {
    float dl[NSUP];
    for (int s = 0; s < NSUP; ++s)
      dl[s] = dist[((size_t)b * NSUP + s) * NQRY + qy];
    for (int r = 0; r < KQ; ++r) {
      float best = 3.402823466e38f; int bi = 0;
      for (int s = 0; s < NSUP; ++s)
        if (dl[s] < best) { best = dl[s]; bi = s; }
      idx[r] = bi;
      dl[bi] = 3.402823466e38f;
    }
  }
  __syncthreads();

  const float* F = feat + (size_t)b * NN * DD;
  float x[KQ + 1];
  x[0] = F[(size_t)(5 + qy) * DD + d];
  for (int r = 0; r < KQ; ++r)
    x[1 + r] = F[(size_t)idx[r] * DD + d];

  float h[16];
  for (int u = 0; u < 16; ++u) {
    float a = bq1[u];
    for (int j = 0; j < KQ + 1; ++j) a += x[j] * Wq1[j * 16 + u];
    h[u] = fmaxf(a, 0.f);
  }
  float lg[KQ + 1];
  float mx = -3.402823466e38f;
  for (int o = 0; o < KQ + 1; ++o) {
    float a = bq2[o];
    for (int u = 0; u < 16; ++u) a += h[u] * Wq2[u * (KQ + 1) + o];
    lg[o] = a; mx = fmaxf(mx, a);
  }
  float sum = 0.f;
  for (int o = 0; o < KQ + 1; ++o) { lg[o] = __expf(lg[o] - mx); sum += lg[o]; }
  float out = 0.f;
  for (int o = 0; o < KQ + 1; ++o) out += x[o] * (lg[o] / sum);

  feature[((size_t)b * NN + 5 + qy) * DD + d] = out;
}

// ---------------------------------------------------------------------------
// Kernel 7: d_all[b,i,j] = sqrt(max(sq[i]+sq[j]-2*G[i,j],0)) via WMMA Gram.
// grid = 16*5*5 blocks of 32.
// ---------------------------------------------------------------------------
__global__ void tcia_cdist_wmma(const float* __restrict__ feature,
                                const float* __restrict__ sqn,
                                float* __restrict__ dall) {
  int b  = blockIdx.x / 25;
  int t  = blockIdx.x % 25;
  int mt = t / 5;
  int nt = t % 5;
  int lane = threadIdx.x;
  int lo = lane & 15;
  int hi = lane >> 4;
  const float* F = feature + (size_t)b * NN * DD;

  int ra = mt * 16 + lo;
  int rb = nt * 16 + lo;

  v8f c = {0.f, 0.f, 0.f, 0.f, 0.f, 0.f, 0.f, 0.f};
  for (int kk = 0; kk < DD; kk += 4) {
    v2f a, w;
    a.x = F[(size_t)ra * DD + kk + 2 * hi];
    a.y = F[(size_t)ra * DD + kk + 2 * hi + 1];
    w.x = F[(size_t)rb * DD + kk + 2 * hi];
    w.y = F[(size_t)rb * DD + kk + 2 * hi + 1];
    c = __builtin_amdgcn_wmma_f32_16x16x4_f32(false, a, false, w,
                                              (short)0, c, false, false);
  }
  for (int v = 0; v < 8; ++v) {
    int m = mt * 16 + v + hi * 8;
    int n = nt * 16 + lo;
    float d2 = sqn[b * NN + m] + sqn[b * NN + n] - 2.f * c[v];
    dall[((size_t)b * NN + m) * NN + n] = sqrtf(fmaxf(d2, 0.f));
  }
}

// ---------------------------------------------------------------------------
// Kernel 8: per-batch triplet loss mean -> lmean[b]. grid = 16, block = 256.
// ---------------------------------------------------------------------------
__global__ void tcia_triplet(const float* __restrict__ dall,
                             const int* __restrict__ label,
                             float* __restrict__ lmean) {
  __shared__ int pos[NSUP][16];
  __shared__ int neg[NSUP][64];
  __shared__ float rsum[256];
  __shared__ float rcnt[256];

  int b = blockIdx.x;
  int t = threadIdx.x;

  if (t < NSUP) {
    int c = t, np = 0, nn = 0;
    for (int j = 0; j < NN; ++j) {
      if (label[j] == c) pos[c][np++] = j;
      else               neg[c][nn++] = j;
    }
  }
  __syncthreads();

  const float* D = dall + (size_t)b * NN * NN;
  float sum = 0.f; float cnt = 0.f;
  const int TOT = NN * 16 * 64;
  for (int it = t; it < TOT; it += 256) {
    int i = it / 1024;
    int r = it % 1024;
    int p = r / 64;
    int n = r % 64;
    int c = label[i];
    float v = MARGIN + D[(size_t)i * NN + pos[c][p]] - D[(size_t)i * NN + neg[c][n]];
    if (v > 0.f) { sum += v; cnt += 1.f; }
  }
  rsum[t] = sum; rcnt[t] = cnt;
  __syncthreads();
  for (int s = 128; s >= 1; s >>= 1) {
    if (t < s) { rsum[t] += rsum[t + s]; rcnt[t] += rcnt[t + s]; }
    __syncthreads();
  }
  if (t == 0) {
    float ln = rcnt[0];
    lmean[b] = (ln == 0.f) ? 0.f : (rsum[0] / fmaxf(ln, 1.f));
  }
}

// ---------------------------------------------------------------------------
// Kernel 9: y_pred = softmax(-mean_b cdist(query_feat, protos)). grid = 75.
// ---------------------------------------------------------------------------
__global__ void tcia_ypred(const float* __restrict__ feature,
                           float* __restrict__ out) {
  __shared__ float red[256];
  __shared__ float dqa[NSUP];
  int qy = blockIdx.x;
  int t  = threadIdx.x;

  for (int s = 0; s < NSUP; ++s) {
    float acc = 0.f;
    for (int b = 0; b < BB; ++b) {
      float diff = feature[((size_t)b * NN + 5 + qy) * DD + t]
                 - feature[((size_t)b * NN + s) * DD + t];
      red[t] = diff * diff;
      __syncthreads();
      for (int ss = 128; ss >= 1; ss >>= 1) {
        if (t < ss) red[t] += red[t + ss];
        __syncthreads();
      }
      if (t == 0) acc += sqrtf(fmaxf(red[0], 0.f));
      __syncthreads();
    }
    if (t == 0) dqa[s] = acc / (float)BB;
    __syncthreads();
  }
  if (t == 0) {
    float mx = -3.402823466e38f;
    for (int s = 0; s < NSUP; ++s) mx = fmaxf(mx, -dqa[s]);
    float sum = 0.f; float e[NSUP];
    for (int s = 0; s < NSUP; ++s) { e[s] = __expf(-dqa[s] - mx); sum += e[s]; }
    for (int s = 0; s < NSUP; ++s) out[qy * NSUP + s] = e[s] / sum;
  }
}

// ---------------------------------------------------------------------------
// Kernel 10: final loss = mean_b lmean[b] -> out[375].
// ---------------------------------------------------------------------------
__global__ void tcia_loss(const float* __restrict__ lmean,
                          float* __restrict__ out) {
  if (threadIdx.x == 0) {
    float s = 0.f;
    for (int b = 0; b < BB; ++b) s += lmean[b];
    out[NQRY * NSUP] = s / (float)BB;
  }
}

// ---------------------------------------------------------------------------
extern "C" void kernel_launch(void* const* d_in, const int* in_sizes, int n_in,
                              void* d_out, int out_size, void* d_ws, size_t ws_size,
                              hipStream_t stream) {
  const float* inpt = (const float*)d_in[0];
  const float* Wqk  = (const float*)d_in[1];
  const float* bqk  = (const float*)d_in[2];
  const float* Ws1  = (const float*)d_in[3];
  const float* bs1  = (const float*)d_in[4];
  const float* Ws2  = (const float*)d_in[5];
  const float* bs2  = (const float*)d_in[6];
  const float* Wq1  = (const float*)d_in[7];
  const float* bq1  = (const float*)d_in[8];
  const float* Wq2  = (const float*)d_in[9];
  const float* bq2  = (const float*)d_in[10];
  const int*   lab  = (const int*)d_in[11];
  float* out = (float*)d_out;

  float* ws = (float*)d_ws;
  const size_t SZ = (size_t)BB * NN * DD;   // 327680
  float* q       = ws;             ws += SZ;
  float* k       = ws;             ws += SZ;
  float* feat    = ws;             ws += SZ;
  float* norms   = ws;             ws += (size_t)BB * NN;
  float* dist    = ws;             ws += (size_t)BB * NSUP * NQRY;
  float* feature = ws;             ws += SZ;
  float* sqn     = ws;             ws += (size_t)BB * NN;
  float* dall    = ws;             ws += (size_t)BB * NN * NN;
  float* lmean   = ws;             ws += BB;

  // 1) qk projection (WMMA f32)
  tcia_qk_wmma<<<dim3(BB * 5 * 32), dim3(32), 0, stream>>>(inpt, Wqk, bqk, q, k);
  // 2) attention softmax + feat
  tcia_attn_feat<<<dim3(BB, DD), dim3(256), 0, stream>>>(inpt, q, k, feat);
  // 3) row norms of feat
  tcia_rownorm<<<dim3(BB * NN), dim3(256), 0, stream>>>(feat, norms, 1);
  // 4) cosine distance (WMMA f32)
  tcia_cos_wmma<<<dim3(BB * 5), dim3(32), 0, stream>>>(feat, norms, dist);
  // 5) support gating -> feature[:, :5]
  tcia_support_mlp<<<dim3(BB * NSUP), dim3(256), 0, stream>>>(
      feat, dist, Ws1, bs1, Ws2, bs2, feature);
  // 6) query gating -> feature[:, 5:]
  tcia_query_mlp<<<dim3(BB * NQRY), dim3(256), 0, stream>>>(
      feat, dist, Wq1, bq1, Wq2, bq2, feature);
  // 7) squared norms of feature rows
  tcia_rownorm<<<dim3(BB * NN), dim3(256), 0, stream>>>(feature, sqn, 0);
  // 8) full pairwise distances (WMMA f32 Gram)
  tcia_cdist_wmma<<<dim3(BB * 25), dim3(32), 0, stream>>>(feature, sqn, dall);
  // 9) triplet loss per batch
  tcia_triplet<<<dim3(BB), dim3(256), 0, stream>>>(dall, lab, lmean);
  // 10) y_pred
  tcia_ypred<<<dim3(NQRY), dim3(256), 0, stream>>>(feature, out);
  // 11) scalar loss
  tcia_loss<<<dim3(1), dim3(32), 0, stream>>>(lmean, out);
}